// GaussianModel_2362232013506
// MI455X (gfx1250) — compile-verified
//
#include <hip/hip_runtime.h>
#include <math.h>
#include <stdint.h>

typedef float v4f __attribute__((ext_vector_type(4)));
typedef int   v4i __attribute__((vector_size(16)));   // matches builtin's param pointee

#define BLK 256

#if defined(__AMDGCN__) && __has_builtin(__builtin_amdgcn_global_load_async_to_lds_b128) && __has_builtin(__builtin_amdgcn_s_wait_asynccnt)
#define USE_ASYNC_LDS 1
#else
#define USE_ASYNC_LDS 0
#endif

#if defined(__AMDGCN__) && __has_builtin(__builtin_amdgcn_rsqf)
#define FAST_RSQ(x) __builtin_amdgcn_rsqf(x)
#else
#define FAST_RSQ(x) (1.0f / sqrtf(x))
#endif

#if defined(__AMDGCN__) && __has_builtin(__builtin_amdgcn_rcpf)
#define FAST_RCP(x) __builtin_amdgcn_rcpf(x)
#else
#define FAST_RCP(x) (1.0f / (x))
#endif

#if defined(__HIP_DEVICE_COMPILE__)
#define FAST_EXP(x) __expf(x)
#else
#define FAST_EXP(x) expf(x)
#endif

// One point per lane; per 256-point tile the stride-12B arrays (xyz, scaling)
// are staged into LDS with 128-bit async global->LDS copies so every global
// transaction is an aligned B128 (or fully covered B32 triplet for xyz_cam).
// All streaming traffic is non-temporal: 272 MB passes once through the
// 192 MB L2 with no reuse. All address math is 32-bit (offsets < 112 MB) so
// the backend can use saddr+voffset global addressing.
__global__ __launch_bounds__(BLK) void splat_cov2d_kernel(
    const float* __restrict__ xyz, const float* __restrict__ scl,
    const float* __restrict__ rot, const float* __restrict__ Rm,
    const float* __restrict__ Tv,  const float* __restrict__ fxp,
    float* __restrict__ out, int n)
{
  __shared__ float lds_xyz[BLK * 3];
  __shared__ float lds_scl[BLK * 3];

  const unsigned tid = threadIdx.x;
  const unsigned un  = (unsigned)n;
  const unsigned p   = blockIdx.x * BLK + tid;

  // ---- stage xyz & scaling tiles into LDS via async B128 copies ----
  const unsigned nf4 = (BLK * 3) / 4;        // 192 float4 per tile
  const unsigned f4max = (3u * un) / 4u;     // total float4 per array (3M)
  if (tid < nf4) {
    unsigned g = blockIdx.x * nf4 + tid;
    if (g >= f4max) g = f4max - 1u;          // clamp: loads stay in-bounds
    const v4f* gx = (const v4f*)xyz + g;
    const v4f* gs = (const v4f*)scl + g;
#if USE_ASYNC_LDS
    typedef __attribute__((address_space(1))) v4i* gp_t;  // global (AS1) int4*
    typedef __attribute__((address_space(3))) v4i* lp_t;  // LDS (AS3) int4*
    __builtin_amdgcn_global_load_async_to_lds_b128(
        (gp_t)(uintptr_t)gx, (lp_t)(uintptr_t)&lds_xyz[tid * 4], 0, 0);
    __builtin_amdgcn_global_load_async_to_lds_b128(
        (gp_t)(uintptr_t)gs, (lp_t)(uintptr_t)&lds_scl[tid * 4], 0, 0);
#else
    *(v4f*)&lds_xyz[tid * 4] = __builtin_nontemporal_load(gx);
    *(v4f*)&lds_scl[tid * 4] = __builtin_nontemporal_load(gs);
#endif
  }
#if USE_ASYNC_LDS
  __builtin_amdgcn_s_wait_asynccnt(0);   // producer waits before signaling
#endif
  __syncthreads();

  if (p >= un) return;

  // ---- uniforms (uniform addresses -> SMEM s_load, broadcast) ----
  const float R00 = Rm[0], R01 = Rm[1], R02 = Rm[2];
  const float R10 = Rm[3], R11 = Rm[4], R12 = Rm[5];
  const float R20 = Rm[6], R21 = Rm[7], R22 = Rm[8];
  const float T0 = Tv[0], T1 = Tv[1], T2 = Tv[2];
  const float f  = fxp[0];

  // ---- quaternion -> rotation (v_rsq normalize) ----
  const v4f q4 = __builtin_nontemporal_load((const v4f*)rot + p);
  float qw = q4.x, qx = q4.y, qy = q4.z, qz = q4.w;
  const float inr = FAST_RSQ(qw * qw + qx * qx + qy * qy + qz * qz);
  qw *= inr; qx *= inr; qy *= inr; qz *= inr;

  const float xx = qx * qx, yy = qy * qy, zz = qz * qz;
  const float xy = qx * qy, xz = qx * qz, yz = qy * qz;
  const float wx = qw * qx, wy = qw * qy, wz = qw * qz;
  const float r00 = 1.f - 2.f * (yy + zz), r01 = 2.f * (xy - wz), r02 = 2.f * (xz + wy);
  const float r10 = 2.f * (xy + wz), r11 = 1.f - 2.f * (xx + zz), r12 = 2.f * (yz - wx);
  const float r20 = 2.f * (xz - wy), r21 = 2.f * (yz + wx), r22 = 1.f - 2.f * (xx + yy);

  // ---- per-point data from LDS (dword index 3*lane: conflict-free, 64 banks) ----
  const float px = lds_xyz[tid * 3 + 0];
  const float py = lds_xyz[tid * 3 + 1];
  const float pz = lds_xyz[tid * 3 + 2];
  const float s0 = fminf(FAST_EXP(lds_scl[tid * 3 + 0]), 10.0f);
  const float s1 = fminf(FAST_EXP(lds_scl[tid * 3 + 1]), 10.0f);
  const float s2 = fminf(FAST_EXP(lds_scl[tid * 3 + 2]), 10.0f);

  // ---- M = Rq * diag(s);  cov3D = M M^T (symmetric) ----
  const float m00 = r00 * s0, m01 = r01 * s1, m02 = r02 * s2;
  const float m10 = r10 * s0, m11 = r11 * s1, m12 = r12 * s2;
  const float m20 = r20 * s0, m21 = r21 * s1, m22 = r22 * s2;

  const float c00 = m00 * m00 + m01 * m01 + m02 * m02;
  const float c01 = m00 * m10 + m01 * m11 + m02 * m12;
  const float c02 = m00 * m20 + m01 * m21 + m02 * m22;
  const float c11 = m10 * m10 + m11 * m11 + m12 * m12;
  const float c12 = m10 * m20 + m11 * m21 + m12 * m22;
  const float c22 = m20 * m20 + m21 * m21 + m22 * m22;

  // ---- camera transform ----
  const float cx = R00 * px + R01 * py + R02 * pz + T0;
  const float cy = R10 * px + R11 * py + R12 * pz + T1;
  const float cz = R20 * px + R21 * py + R22 * pz + T2;

  const float zed  = cz + 1e-7f;
  const float rz   = FAST_RCP(zed);
  const float invz = f * rz;            // f / z
  const float inz2 = -invz * rz;        // -f / z^2
  const float j02  = cx * inz2;         // -(f*x)/z^2
  const float j12  = cy * inz2;         // -(f*y)/z^2

  // ---- U = J @ R (2x3) ----
  const float u00 = invz * R00 + j02 * R20;
  const float u01 = invz * R01 + j02 * R21;
  const float u02 = invz * R02 + j02 * R22;
  const float u10 = invz * R10 + j12 * R20;
  const float u11 = invz * R11 + j12 * R21;
  const float u12 = invz * R12 + j12 * R22;

  // ---- cov2d = U cov3D U^T + 0.3 I  (cov3D symmetric => o10 == o01) ----
  const float w00 = u00 * c00 + u01 * c01 + u02 * c02;
  const float w01 = u00 * c01 + u01 * c11 + u02 * c12;
  const float w02 = u00 * c02 + u01 * c12 + u02 * c22;
  const float w10 = u10 * c00 + u11 * c01 + u12 * c02;
  const float w11 = u10 * c01 + u11 * c11 + u12 * c12;
  const float w12 = u10 * c02 + u11 * c12 + u12 * c22;

  const float o00 = w00 * u00 + w01 * u01 + w02 * u02 + 0.3f;
  const float o01 = w00 * u10 + w01 * u11 + w02 * u12;
  const float o11 = w10 * u10 + w11 * u11 + w12 * u12 + 0.3f;

  // ---- streaming NT stores (32-bit offsets) ----
  const v4f o = {o00, o01, o01, o11};
  __builtin_nontemporal_store(o, (v4f*)out + p);

  float* cam = out + (4u * un + 3u * p);
  __builtin_nontemporal_store(cx, cam + 0);
  __builtin_nontemporal_store(cy, cam + 1);
  __builtin_nontemporal_store(cz, cam + 2);
}

extern "C" void kernel_launch(void* const* d_in, const int* in_sizes, int n_in,
                              void* d_out, int out_size, void* d_ws, size_t ws_size,
                              hipStream_t stream) {
  (void)n_in; (void)out_size; (void)d_ws; (void)ws_size;
  const float* xyz = (const float*)d_in[0];
  const float* scl = (const float*)d_in[1];
  const float* rot = (const float*)d_in[2];
  const float* Rm  = (const float*)d_in[3];
  const float* Tv  = (const float*)d_in[4];
  const float* fx  = (const float*)d_in[5];
  const int n = in_sizes[0] / 3;           // xyz is (N,3)
  const int blocks = (n + BLK - 1) / BLK;  // N=4e6 -> 15625 full blocks
  splat_cov2d_kernel<<<blocks, BLK, 0, stream>>>(xyz, scl, rot, Rm, Tv, fx,
                                                 (float*)d_out, n);
}